// MultiHeadAttention_13082470383831
// MI455X (gfx1250) — compile-verified
//
#include <hip/hip_runtime.h>

// Problem dims (fixed by reference)
#define BATCH   32
#define TOKENS  2048
#define DDIM    1024      // d_token == n_heads*d_head == 1024
#define TCHUNKS 32
#define TOK_PER_CHUNK (TOKENS / TCHUNKS)   // 64

typedef float v2f __attribute__((ext_vector_type(2)));
typedef float v4f __attribute__((ext_vector_type(4)));
typedef float v8f __attribute__((ext_vector_type(8)));

// ---------------------------------------------------------------------------
// Stage 1: partial token sum.  grid = BATCH*TCHUNKS blocks, 256 threads.
// Each thread owns 4 consecutive channels; block reads 4KB contiguous per
// token step -> fully coalesced streaming of the 256 MiB x tensor.
// ---------------------------------------------------------------------------
__global__ void token_partial_sum(const float* __restrict__ x,
                                  float* __restrict__ partials) {
    const int blk = blockIdx.x;
    const int b  = blk / TCHUNKS;
    const int tc = blk % TCHUNKS;
    const int d0 = threadIdx.x * 4;

    const float* xp = x + ((size_t)b * TOKENS + (size_t)tc * TOK_PER_CHUNK) * DDIM + d0;

    v4f acc = {0.f, 0.f, 0.f, 0.f};
#pragma unroll 4
    for (int t = 0; t < TOK_PER_CHUNK; ++t) {
        v4f v = __builtin_nontemporal_load((const v4f*)(xp + (size_t)t * DDIM));
        acc += v;
    }
    *(v4f*)(partials + (size_t)blk * DDIM + d0) = acc;
}

// ---------------------------------------------------------------------------
// Stage 2: fold TCHUNKS partials into xs[b][d].  grid = BATCH, 256 threads.
// ---------------------------------------------------------------------------
__global__ void token_final_sum(const float* __restrict__ partials,
                                float* __restrict__ xs) {
    const int b  = blockIdx.x;
    const int d0 = threadIdx.x * 4;
    v4f acc = {0.f, 0.f, 0.f, 0.f};
#pragma unroll
    for (int tc = 0; tc < TCHUNKS; ++tc) {
        acc += *(const v4f*)(partials + ((size_t)(b * TCHUNKS + tc)) * DDIM + d0);
    }
    *(v4f*)(xs + (size_t)b * DDIM + d0) = acc;
}

// ---------------------------------------------------------------------------
// WMMA GEMM:  C[M=32, N=1024] = A[32, K=1024] * W[N=1024, K=1024]^T (+ bias)
// One 16x16 output tile per wave via V_WMMA_F32_16X16X4_F32 (exact f32 MAC).
// A frag (16x4): lane m = lane%16, VGPR v holds K = (lane>>4)*2 + v.
// B frag (4x16): lane n = lane%16, VGPR v holds K = (lane>>4)*2 + v.
// C/D (16x16):   N = lane%16, VGPR v holds M = v + (lane>>4)*8.
// grid = 32 blocks * 4 waves = 128 tiles (2 m-tiles x 64 n-tiles).
// ---------------------------------------------------------------------------
__global__ void gemm32_wmma_f32(const float* __restrict__ A,
                                const float* __restrict__ W,
                                const float* __restrict__ bias,
                                float* __restrict__ C,
                                int useBias) {
    const int K = 1024, N = 1024;
    const int lane  = threadIdx.x & 31;
    const int wave  = threadIdx.x >> 5;
    const int tile  = blockIdx.x * 4 + wave;   // 0..127
    const int mtile = tile & 1;                // 0..1
    const int ntile = tile >> 1;               // 0..63

    const int m  = mtile * 16 + (lane & 15);
    const int n  = ntile * 16 + (lane & 15);
    const int kb = (lane >> 4) * 2;            // 0 or 2

    const float* arow = A + (size_t)m * K + kb;
    const float* brow = W + (size_t)n * K + kb;   // W is [N,K] row-major => B^T rows

    v8f acc = {};
#pragma unroll 8
    for (int ks = 0; ks < K; ks += 4) {
        v2f a = *(const v2f*)(arow + ks);
        v2f b = *(const v2f*)(brow + ks);
        acc = __builtin_amdgcn_wmma_f32_16x16x4_f32(
            /*neg_a=*/false, a, /*neg_b=*/false, b,
            /*c_mod=*/(short)0, acc, /*reuse_a=*/false, /*reuse_b=*/false);
    }

    const float bv = useBias ? bias[n] : 0.0f;
    const int mbase = mtile * 16 + (lane >> 4) * 8;
#pragma unroll
    for (int v = 0; v < 8; ++v) {
        C[(size_t)(mbase + v) * N + n] = acc[v] + bv;
    }
}

// ---------------------------------------------------------------------------
// Launch.  ws layout: [partials: B*TCHUNKS*D][xs: B*D][xcat: B*D]  (~4.25 MB)
// ---------------------------------------------------------------------------
extern "C" void kernel_launch(void* const* d_in, const int* in_sizes, int n_in,
                              void* d_out, int out_size, void* d_ws, size_t ws_size,
                              hipStream_t stream) {
    const float* x       = (const float*)d_in[0]; // [32,2048,1024]
    const float* W_heads = (const float*)d_in[1]; // [16,64,1024] == [1024,1024] row-major
    const float* W_proj  = (const float*)d_in[2]; // [1024,1024]  row-major [out,in]
    const float* b_proj  = (const float*)d_in[3]; // [1024]
    float* out = (float*)d_out;                   // [32,1024]

    float* partials = (float*)d_ws;                             // 32*32*1024
    float* xs       = partials + (size_t)BATCH * TCHUNKS * DDIM; // 32*1024
    float* xcat     = xs + (size_t)BATCH * DDIM;                 // 32*1024

    token_partial_sum<<<BATCH * TCHUNKS, 256, 0, stream>>>(x, partials);
    token_final_sum<<<BATCH, 256, 0, stream>>>(partials, xs);
    gemm32_wmma_f32<<<32, 128, 0, stream>>>(xs, W_heads, nullptr, xcat, 0);
    gemm32_wmma_f32<<<32, 128, 0, stream>>>(xcat, W_proj, b_proj, out, 1);
}